// CRFRNN_71588514890336
// MI455X (gfx1250) — compile-verified
//
#include <hip/hip_runtime.h>
#include <stdint.h>

// -------- problem constants (from reference) --------
#define BB 8
#define CC 4
#define H_IMG 64
#define W_IMG 2048
#define HW_IMG (H_IMG * W_IMG)          // 131072
#define KH 3
#define KW 5
#define TH 16
#define TW 64
#define HALO_H (TH + KH - 1)            // 18
#define HALO_W (TW + KW - 1)            // 68
#define NHALO (HALO_H * HALO_W)         // 1224
#define NEG_INV_2B2 (-2222.2222222f)    // -1/(2*0.015^2)

typedef __attribute__((ext_vector_type(2))) float v2f;
typedef __attribute__((ext_vector_type(8))) float v8f;
typedef __attribute__((ext_vector_type(4))) unsigned int u32x4;
typedef __attribute__((ext_vector_type(8))) int i32x8;
typedef __attribute__((ext_vector_type(4))) int i32x4;

#if __has_builtin(__builtin_amdgcn_tensor_load_to_lds) && __has_builtin(__builtin_amdgcn_s_wait_tensorcnt)
#define HAVE_TDM 1
#else
#define HAVE_TDM 0
#endif
#if __has_builtin(__builtin_amdgcn_wmma_f32_16x16x4_f32)
#define HAVE_WMMA 1
#else
#define HAVE_WMMA 0
#endif

// theta_gamma == theta_alpha == 0.9  ->  one shared 3x5 Gaussian, center zeroed.
// g = exp(-(dh^2+dw^2)/(2*0.81))
__constant__ float GK[KH][KW] = {
    {0.0456651f, 0.2909607f, 0.5394071f, 0.2909607f, 0.0456651f},
    {0.0846581f, 0.5394071f, 0.0f,       0.5394071f, 0.0846581f},
    {0.0456651f, 0.2909607f, 0.5394071f, 0.2909607f, 0.0456651f}};

__device__ __forceinline__ float4 softmax4(float a, float b, float c, float d) {
  float m  = fmaxf(fmaxf(a, b), fmaxf(c, d));
  float e0 = __expf(a - m), e1 = __expf(b - m), e2 = __expf(c - m), e3 = __expf(d - m);
  float r  = 1.0f / (e0 + e1 + e2 + e3);
  return make_float4(e0 * r, e1 * r, e2 * r, e3 * r);
}

__global__ __launch_bounds__(256)
void crf_iter(const float* __restrict__ Qin, const float* __restrict__ unary,
              const float* __restrict__ xyz, const float* __restrict__ mask,
              const float* __restrict__ wapp, const float* __restrict__ wsmo,
              const float* __restrict__ compat, float* __restrict__ Qout) {
  // LDS: TDM target (c-major), softmaxed probs (pixel-major float4),
  // packed {x,y,z,mask}, per-wave WMMA staging.  Total ~61.4 KB.
  __shared__ float  Qraw[CC][HALO_H][HALO_W];
  __shared__ float4 Pf[HALO_H][HALO_W];
  __shared__ float4 XM[HALO_H][HALO_W];
  __shared__ float4 wkbuf[8][32];

  const int tid  = threadIdx.x;
  const int lane = tid & 31;
  const int wave = tid >> 5;
  const int bx = blockIdx.x, by = blockIdx.y, b = blockIdx.z;
  const int h0 = by * TH, w0 = bx * TW;

  const float* Qb = Qin + (size_t)b * CC * HW_IMG;

  // Interior tile: entire halo is inside the image -> TDM fast path (no OOB).
  const bool interior = (by >= 1) && (by <= (H_IMG / TH - 2)) &&
                        (bx >= 1) && (bx <= (W_IMG / TW - 2));
#if HAVE_TDM
  const bool useTDM = interior;
#else
  const bool useTDM = false;
#endif

#if HAVE_TDM
  if (useTDM && wave == 0) {
    // ---- Tensor DMA descriptor (ISA cdna5 8.3-8.6), 3D tile: ----
    //   dims (x,y,z) = (W, H, C); strides = (W, H*W); tile = (68, 18, 4), f32.
    unsigned long long ga =
        (unsigned long long)(uintptr_t)Qb +
        4ull * ((unsigned long long)(h0 - 1) * W_IMG + (unsigned)(w0 - 2));
    unsigned ldsOff = (unsigned)(uintptr_t)&Qraw[0][0][0];  // low 32b = LDS byte offset
    u32x4 g0;
    g0[0] = 1u;                                       // count=1, user mode
    g0[1] = ldsOff;                                   // lds_addr
    g0[2] = (unsigned)(ga & 0xFFFFFFFFull);           // global_addr[31:0]
    g0[3] = (unsigned)((ga >> 32) & 0x01FFFFFFull)    // global_addr[56:32]
            | 0x80000000u;                            // type=2 ("image")
    i32x8 g1;
    g1[0] = 0x00020000;                               // data_size=4B; no mcast/pad
    g1[1] = (int)((unsigned)W_IMG << 16);             // tensor_dim0 lo16
    g1[2] = (int)(((unsigned)W_IMG >> 16) | ((unsigned)H_IMG << 16));
    g1[3] = (int)(((unsigned)H_IMG >> 16) | ((unsigned)HALO_W << 16)); // tile_dim0=68
    g1[4] = (int)((unsigned)HALO_H | ((unsigned)CC << 16));            // tile_dim1/2
    g1[5] = W_IMG;                                    // tensor_dim0_stride lo32
    g1[6] = 0;                                        // stride0 hi | stride1 lo16 (=0)
    g1[7] = (int)((unsigned)HW_IMG >> 16);            // stride1[47:16] = 2
    i32x4 g2; g2[0] = CC; g2[1] = 0; g2[2] = 0; g2[3] = 0;  // tensor_dim2=4
    i32x4 g3; g3[0] = 0;  g3[1] = 0; g3[2] = 0; g3[3] = 0;
    i32x8 g4;                                          // extra operand of the
    g4[0] = 0; g4[1] = 0; g4[2] = 0; g4[3] = 0;        // clang-23 6-arg builtin
    g4[4] = 0; g4[5] = 0; g4[6] = 0; g4[7] = 0;        // (zero-filled)
    __builtin_amdgcn_tensor_load_to_lds(g0, g1, g2, g3, g4, 0);
  }
#endif

  // ---- Stage A: fill XM (xyz+mask) and, on the guarded path, Pf ----
  for (int i = tid; i < NHALO; i += 256) {
    int hh = i / HALO_W, ww = i - hh * HALO_W;
    int gh = h0 - 1 + hh, gw = w0 - 2 + ww;
    bool inb = (gh >= 0) && (gh < H_IMG) && (gw >= 0) && (gw < W_IMG);
    float4 v = make_float4(0.f, 0.f, 0.f, 0.f);
    if (inb) {
      size_t off = (size_t)gh * W_IMG + gw;
      const float* xb = xyz + (size_t)b * 3 * HW_IMG + off;
      v.x = xb[0]; v.y = xb[HW_IMG]; v.z = xb[2 * HW_IMG];
      v.w = mask[(size_t)b * HW_IMG + off];
    }
    XM[hh][ww] = v;
    if (!useTDM) {
      float4 pv = make_float4(0.f, 0.f, 0.f, 0.f);  // zero-pad post-softmax Q
      if (inb) {
        size_t off = (size_t)gh * W_IMG + gw;
        const float* qp = Qb + off;
        pv = softmax4(qp[0], qp[HW_IMG], qp[2 * HW_IMG], qp[3 * HW_IMG]);
      }
      Pf[hh][ww] = pv;
    }
  }
#if HAVE_TDM
  if (useTDM) {
    if (wave == 0) __builtin_amdgcn_s_wait_tensorcnt(0);
    __syncthreads();  // Qraw visible to all waves
    for (int i = tid; i < NHALO; i += 256) {
      int hh = i / HALO_W, ww = i - hh * HALO_W;
      Pf[hh][ww] = softmax4(Qraw[0][hh][ww], Qraw[1][hh][ww],
                            Qraw[2][hh][ww], Qraw[3][hh][ww]);
    }
  }
#endif
  __syncthreads();

  // per-channel weights (broadcast scalars)
  const float wa0 = wapp[0], wa1 = wapp[1], wa2 = wapp[2], wa3 = wapp[3];
  const float ws0 = wsmo[0], ws1 = wsmo[1], ws2 = wsmo[2], ws3 = wsmo[3];

#if HAVE_WMMA
  // A operand: compat_w (4x4) zero-padded to 16x4.  f32 A layout (ISA 7.12.2):
  // lane L<16 holds A[M=L][K=0..1]; lane L>=16 holds A[M=L-16][K=2..3].
  v2f av;
  {
    int m = lane & 15, kh = lane >> 4;
    if (m < CC) { av.x = compat[m * CC + 2 * kh]; av.y = compat[m * CC + 2 * kh + 1]; }
    else        { av.x = 0.f; av.y = 0.f; }
  }
#endif

  // ---- Stage B: 4 rounds x 32 pixels per wave (tile = 8 waves * 128 px) ----
  const int waveBase = wave * 128;
#pragma unroll 1
  for (int r = 0; r < 4; ++r) {
    int pix = waveBase + r * 32 + lane;
    int ph = pix >> 6, pw = pix & 63;          // local tile coords
    float4 ctr = XM[ph + 1][pw + 2];
    float ks0 = 0.f, ks1 = 0.f, ks2 = 0.f, ks3 = 0.f;   // = k_smooth = exp_smooth
    float ea0 = 0.f, ea1 = 0.f, ea2 = 0.f, ea3 = 0.f;   // bilateral accumulator
#pragma unroll
    for (int dh = 0; dh < KH; ++dh) {
#pragma unroll
      for (int dw = 0; dw < KW; ++dw) {
        if (dh == 1 && dw == 2) continue;      // center tap zeroed
        float g = GK[dh][dw];
        float4 nx = XM[ph + dh][pw + dw];
        float dx = nx.x - ctr.x, dy = nx.y - ctr.y, dz = nx.z - ctr.z;
        float pd = fmaf(dx, dx, fmaf(dy, dy, dz * dz));
        float kb = __expf(pd * NEG_INV_2B2) * nx.w;     // kernel_beta * m_nb
        float4 p = Pf[ph + dh][pw + dw];
        ks0 = fmaf(g, p.x, ks0); ks1 = fmaf(g, p.y, ks1);
        ks2 = fmaf(g, p.z, ks2); ks3 = fmaf(g, p.w, ks3);
        ea0 = fmaf(kb, p.x, ea0); ea1 = fmaf(kb, p.y, ea1);
        ea2 = fmaf(kb, p.z, ea2); ea3 = fmaf(kb, p.w, ea3);
      }
    }
    // weighted_k = ws*k_smooth + wa*(m*exp_app)*exp_smooth   (k_smooth==exp_smooth)
    float mc = ctr.w;
    float wk0 = fmaf(wa0 * (ea0 * mc), ks0, ws0 * ks0);
    float wk1 = fmaf(wa1 * (ea1 * mc), ks1, ws1 * ks1);
    float wk2 = fmaf(wa2 * (ea2 * mc), ks2, ws2 * ks2);
    float wk3 = fmaf(wa3 * (ea3 * mc), ks3, ws3 * ks3);

#if HAVE_WMMA
    // Stage weighted_k to LDS, then compat mix via V_WMMA_F32_16X16X4_F32 (K=4==C).
    wkbuf[wave][lane] = make_float4(wk0, wk1, wk2, wk3);
    int n = lane & 15, kh = lane >> 4;
    const float2* bp0 = (const float2*)&wkbuf[wave][n];
    const float2* bp1 = (const float2*)&wkbuf[wave][16 + n];
    float2 t0 = bp0[kh], t1 = bp1[kh];
    v2f bv0; bv0.x = t0.x; bv0.y = t0.y;
    v2f bv1; bv1.x = t1.x; bv1.y = t1.y;
    v8f cz = {0.f, 0.f, 0.f, 0.f, 0.f, 0.f, 0.f, 0.f};
    v8f d0 = __builtin_amdgcn_wmma_f32_16x16x4_f32(false, av, false, bv0,
                                                   (short)0, cz, false, false);
    v8f d1 = __builtin_amdgcn_wmma_f32_16x16x4_f32(false, av, false, bv1,
                                                   (short)0, cz, false, false);
    // D layout: lane L<16, VGPR j -> D[M=j][N=L]; output channels are rows 0..3.
    if (lane < 16) {
      int p0 = waveBase + r * 32 + lane;
      int p1 = p0 + 16;
      int gh0 = h0 + (p0 >> 6), gw0 = w0 + (p0 & 63);
      int gh1 = h0 + (p1 >> 6), gw1 = w0 + (p1 & 63);
#pragma unroll
      for (int c = 0; c < CC; ++c) {
        size_t i0 = ((size_t)(b * CC + c) * H_IMG + gh0) * W_IMG + gw0;
        size_t i1 = ((size_t)(b * CC + c) * H_IMG + gh1) * W_IMG + gw1;
        Qout[i0] = unary[i0] - d0[c];
        Qout[i1] = unary[i1] - d1[c];
      }
    }
#else
    // VALU fallback: pairwise[o] = sum_c compat[o][c] * wk[c]
    float wk[4] = {wk0, wk1, wk2, wk3};
    int gh = h0 + ph, gw = w0 + pw;
#pragma unroll
    for (int o = 0; o < CC; ++o) {
      float pwv = 0.f;
#pragma unroll
      for (int c = 0; c < CC; ++c) pwv = fmaf(compat[o * CC + c], wk[c], pwv);
      size_t i0 = ((size_t)(b * CC + o) * H_IMG + gh) * W_IMG + gw;
      Qout[i0] = unary[i0] - pwv;
    }
#endif
  }
}

extern "C" void kernel_launch(void* const* d_in, const int* in_sizes, int n_in,
                              void* d_out, int out_size, void* d_ws, size_t ws_size,
                              hipStream_t stream) {
  (void)in_sizes; (void)n_in; (void)out_size; (void)ws_size;
  const float* unary  = (const float*)d_in[0];
  const float* xyz    = (const float*)d_in[1];
  const float* mask   = (const float*)d_in[2];
  const float* wapp   = (const float*)d_in[3];
  const float* wsmo   = (const float*)d_in[4];
  const float* compat = (const float*)d_in[5];
  float* out = (float*)d_out;
  float* ws0 = (float*)d_ws;
  float* ws1 = ws0 + (size_t)BB * CC * HW_IMG;   // ping-pong (2 x 16.8 MB)

  dim3 grid(W_IMG / TW, H_IMG / TH, BB), block(256);
  // Q0 = unary; each launch applies softmax + message passing + compat + update.
  crf_iter<<<grid, block, 0, stream>>>(unary, unary, xyz, mask, wapp, wsmo, compat, ws0);
  crf_iter<<<grid, block, 0, stream>>>(ws0,   unary, xyz, mask, wapp, wsmo, compat, ws1);
  crf_iter<<<grid, block, 0, stream>>>(ws1,   unary, xyz, mask, wapp, wsmo, compat, out);
}